// SparseGrid_56856777064583
// MI455X (gfx1250) — compile-verified
//
#include <hip/hip_runtime.h>
#include <hip/hip_bf16.h>

#define RESO 256
#define NCH  27   // BASIS_DIM * 3

__global__ __launch_bounds__(256) void SparseGrid_sample_kernel(
    const float* __restrict__ points,
    const float* __restrict__ density_data,
    const float* __restrict__ sh_data,
    const int*   __restrict__ links,
    float*       __restrict__ out_density,
    float*       __restrict__ out_sh,
    int n)
{
    int p = blockIdx.x * blockDim.x + threadIdx.x;
    if (p >= n) return;

    const float scaling = 0.5f * (float)RESO;          // 128.0
    const float offs    = 0.5f * (float)RESO - 0.5f;   // 127.5

    float px = points[3 * p + 0];
    float py = points[3 * p + 1];
    float pz = points[3 * p + 2];

    float fx = fminf(fmaxf(fmaf(px, scaling, offs), 0.0f), (float)(RESO - 1));
    float fy = fminf(fmaxf(fmaf(py, scaling, offs), 0.0f), (float)(RESO - 1));
    float fz = fminf(fmaxf(fmaf(pz, scaling, offs), 0.0f), (float)(RESO - 1));

    float lxf = fminf(floorf(fx), (float)(RESO - 2));
    float lyf = fminf(floorf(fy), (float)(RESO - 2));
    float lzf = fminf(floorf(fz), (float)(RESO - 2));

    float tx = fx - lxf, ty = fy - lyf, tz = fz - lzf;
    int lx = (int)lxf, ly = (int)lyf, lz = (int)lzf;

    int base = (lx * RESO + ly) * RESO + lz;

    // Gather the 8 corner links. links has ~15x L2 line reuse -> regular
    // (RT) loads, keep it resident in the 192MB L2.
    int lnk[8];
#pragma unroll
    for (int c = 0; c < 8; ++c) {
        int dx = (c >> 2) & 1, dy = (c >> 1) & 1, dz = c & 1;
        lnk[c] = links[base + dx * RESO * RESO + dy * RESO + dz];
    }

    // global_prefetch_b8 both cachelines of each valid SH row (rows are
    // 108B, arbitrary alignment) so the 8 dependent gathers overlap.
    // locality 0: non-temporal flavor, these lines are used exactly once.
#pragma unroll
    for (int c = 0; c < 8; ++c) {
        if (lnk[c] >= 0) {
            const char* row = (const char*)(sh_data + (size_t)lnk[c] * NCH);
            __builtin_prefetch(row, 0, 0);
            __builtin_prefetch(row + 104, 0, 0);
        }
    }

    float wx[2] = {1.0f - tx, tx};
    float wy[2] = {1.0f - ty, ty};
    float wz[2] = {1.0f - tz, tz};

    float accd = 0.0f;
    float acc[NCH];
#pragma unroll
    for (int k = 0; k < NCH; ++k) acc[k] = 0.0f;

#pragma unroll
    for (int c = 0; c < 8; ++c) {
        if (lnk[c] >= 0) {   // EXEC-masked: invalid corners generate no traffic
            float w = wx[(c >> 2) & 1] * wy[(c >> 1) & 1] * wz[c & 1];
            const float* row = sh_data + (size_t)lnk[c] * NCH;
            // sh_data (~950MB) and density_data have ~1 touch per row:
            // non-temporal so they don't evict the hot links table from L2.
            float d = __builtin_nontemporal_load(density_data + lnk[c]);
            accd = fmaf(w, d, accd);
#pragma unroll
            for (int k = 0; k < NCH; ++k) {
                float v = __builtin_nontemporal_load(row + k);
                acc[k] = fmaf(w, v, acc[k]);
            }
        }
    }

    // Streaming output (224MB total): non-temporal stores.
    __builtin_nontemporal_store(accd, &out_density[p]);
    float* o = out_sh + (size_t)p * NCH;
#pragma unroll
    for (int k = 0; k < NCH; ++k)
        __builtin_nontemporal_store(acc[k], &o[k]);
}

extern "C" void kernel_launch(void* const* d_in, const int* in_sizes, int n_in,
                              void* d_out, int out_size, void* d_ws, size_t ws_size,
                              hipStream_t stream)
{
    const float* points       = (const float*)d_in[0];
    const float* density_data = (const float*)d_in[1];
    const float* sh_data      = (const float*)d_in[2];
    const int*   links        = (const int*)d_in[3];

    int n = in_sizes[0] / 3;                  // 2,000,000 points
    float* out_density = (float*)d_out;       // first n floats
    float* out_sh      = (float*)d_out + n;   // then n * 27 floats

    dim3 block(256);
    dim3 grid((n + 255) / 256);
    SparseGrid_sample_kernel<<<grid, block, 0, stream>>>(
        points, density_data, sh_data, links, out_density, out_sh, n);
}